// SimpleSSM_29128468201575
// MI455X (gfx1250) — compile-verified
//
#include <hip/hip_runtime.h>
#include <hip/hip_bf16.h>

// SimpleSSM on MI455X (gfx1250): memory-bound diagonal linear recurrence.
//   Bu = u @ W_B^T + b_B       -> WMMA f32 16x16x4, K-accumulated over D=32
//   h_t = A*h_{t-1} + Bu_t     -> chunked parallel scan (64 chunks of 128)
//   y_t = W_C h_t + b_C        -> per-t 16-lane reduction + linear correction

#define TT      8192
#define NBATCH  128
#define DIN     32
#define HID     16
#define CHLEN   128               // chunk length L
#define NCHUNK  (TT / CHLEN)      // 64
#define NTILE   (CHLEN / 16)      // 8 WMMA tiles per chunk
#define LDS_STRIDE 17             // padded row stride (dwords) to avoid bank conflicts

typedef float v2f __attribute__((ext_vector_type(2)));
typedef float v8f __attribute__((ext_vector_type(8)));

// ---------------------------------------------------------------------------
// Kernel 1: per (batch, chunk) workgroup (1 wave of 32). WMMA projection of a
// 16-timestep x 32-dim u tile against W_B^T, local scan with h0=0, partial y
// written to d_out, chunk-final state written to workspace S.
// ---------------------------------------------------------------------------
__global__ __launch_bounds__(32) void ssm_chunk_kernel(
    const float* __restrict__ u, const float* __restrict__ A,
    const float* __restrict__ W_B, const float* __restrict__ b_B,
    const float* __restrict__ W_C, const float* __restrict__ b_C,
    float* __restrict__ y_partial, float* __restrict__ S)
{
    __shared__ float buLds[CHLEN * LDS_STRIDE];
    __shared__ float ybuf[CHLEN];

    const int wg = blockIdx.x;
    const int b  = wg / NCHUNK;
    const int c  = wg % NCHUNK;
    const int t0 = c * CHLEN;
    const int ln = threadIdx.x;       // 0..31
    const int m  = ln & 15;           // row (A-frag) / col (B-frag) index
    const int kh = ln >> 4;           // K-half select (lanes 16-31 hold K=2,3)

    // --- B fragments: W_B^T 4x16 slices, resident in VGPRs for whole chunk.
    // B-frag VGPR0 = K row (4k + 2*kh), VGPR1 = K row (4k + 2*kh + 1), col = m.
    // W_B^T[d][h] = W_B[h][d], so the pair is contiguous along d in row h=m.
    v2f bfrag[8];
#pragma unroll
    for (int k = 0; k < 8; ++k) {
        const float* p = W_B + m * DIN + 4 * k + 2 * kh;
        bfrag[k].x = p[0];
        bfrag[k].y = p[1];
    }
    const float bias = b_B[m];

    const float* ubase = u + ((size_t)b * TT + t0) * DIN;

    // --- Phase A: 8 tiles of 16 timesteps, each = 8 chained WMMAs over K=32.
    for (int tile = 0; tile < NTILE; ++tile) {
        // prefetch next tile's u block (emits global_prefetch_b8)
        if (tile + 1 < NTILE)
            __builtin_prefetch(ubase + (tile + 1) * 16 * DIN + ln * 16, 0, 1);

        v8f cacc;
#pragma unroll
        for (int v = 0; v < 8; ++v) cacc[v] = bias;   // accumulator = bias splat

        const float* arow = ubase + (tile * 16 + m) * DIN + 2 * kh;
#pragma unroll
        for (int k = 0; k < 8; ++k) {
            // A-frag: row M=m (timestep), pair of dims (4k+2kh, 4k+2kh+1)
            v2f afrag;
            afrag.x = arow[4 * k];
            afrag.y = arow[4 * k + 1];
            cacc = __builtin_amdgcn_wmma_f32_16x16x4_f32(
                /*neg_a=*/false, afrag, /*neg_b=*/false, bfrag[k],
                /*c_mod=*/(short)0, cacc, /*reuse_a=*/false, /*reuse_b=*/false);
        }

        // C layout: lane ln holds (M = v + 8*kh, N = m) in cacc[v]
#pragma unroll
        for (int v = 0; v < 8; ++v) {
            int t = tile * 16 + v + 8 * kh;
            buLds[t * LDS_STRIDE + m] = cacc[v];
        }
    }

    __syncthreads();

    // --- Phase B: local scan, lanes 0..15 = hidden channels.
    const float bC = b_C[0];
    if (ln < 16) {
        const float Aval = A[ln];
        const float wc   = W_C[ln];
        float hreg = 0.0f;
        for (int t = 0; t < CHLEN; ++t) {
            float bu = buLds[t * LDS_STRIDE + ln];
            hreg = fmaf(Aval, hreg, bu);
            float val = wc * hreg;
            val += __shfl_xor(val, 1);
            val += __shfl_xor(val, 2);
            val += __shfl_xor(val, 4);
            val += __shfl_xor(val, 8);
            if (ln == 0) ybuf[t] = val + bC;
        }
        // chunk-final local state
        S[((size_t)b * NCHUNK + c) * HID + ln] = hreg;
    }

    __syncthreads();

    // --- Phase C: coalesced partial-y store.
    for (int i = ln; i < CHLEN; i += 32)
        y_partial[(size_t)b * TT + t0 + i] = ybuf[i];
}

// ---------------------------------------------------------------------------
// Kernel 2: sequential scan over chunk states per (b, h): F_c = A^L F_{c-1} + S_c.
// Carry[b,c] = F_{c-1} (state entering chunk c). 128 blocks x 32 threads.
// ---------------------------------------------------------------------------
__global__ __launch_bounds__(32) void ssm_carry_kernel(
    const float* __restrict__ A, const float* __restrict__ S,
    float* __restrict__ Carry)
{
    const int b = blockIdx.x;
    const int h = threadIdx.x;
    if (h >= HID) return;

    float a = A[h];
    float p = a;                    // a^128 via 7 squarings (128 = 2^7)
#pragma unroll
    for (int i = 0; i < 7; ++i) p *= p;

    float f = 0.0f;
    for (int c = 0; c < NCHUNK; ++c) {
        size_t idx = ((size_t)b * NCHUNK + c) * HID + h;
        Carry[idx] = f;
        f = fmaf(p, f, S[idx]);
    }
}

// ---------------------------------------------------------------------------
// Kernel 3: linear correction. Global h_t = local h_t + A^{i+1} * carry, so
// y[b,t] += sum_h W_C[h] * A[h]^{i+1} * Carry[b,c,h].  (b*NCHUNK) blocks x L.
// ---------------------------------------------------------------------------
__global__ __launch_bounds__(CHLEN) void ssm_fix_kernel(
    const float* __restrict__ A, const float* __restrict__ W_C,
    const float* __restrict__ Carry, float* __restrict__ y)
{
    const int wg = blockIdx.x;
    const int b  = wg / NCHUNK;
    const int c  = wg % NCHUNK;
    const int i  = threadIdx.x;     // 0..CHLEN-1

    __shared__ float ch[HID], ah[HID], wch[HID];
    if (i < HID) {
        ch[i]  = Carry[((size_t)b * NCHUNK + c) * HID + i];
        ah[i]  = A[i];
        wch[i] = W_C[i];
    }
    __syncthreads();

    float corr = 0.0f;
#pragma unroll
    for (int h = 0; h < HID; ++h)
        corr = fmaf(wch[h] * ch[h], powf(ah[h], (float)(i + 1)), corr);

    y[(size_t)b * TT + (size_t)c * CHLEN + i] += corr;
}

// ---------------------------------------------------------------------------
extern "C" void kernel_launch(void* const* d_in, const int* in_sizes, int n_in,
                              void* d_out, int out_size, void* d_ws, size_t ws_size,
                              hipStream_t stream) {
    const float* u   = (const float*)d_in[0];   // [B, T, D]
    const float* A   = (const float*)d_in[1];   // [H]
    const float* W_B = (const float*)d_in[2];   // [H, D]
    const float* b_B = (const float*)d_in[3];   // [H]
    const float* W_C = (const float*)d_in[4];   // [1, H]
    const float* b_C = (const float*)d_in[5];   // [1]
    float* y = (float*)d_out;                   // [B, T]

    // workspace: chunk-final states S then carries, each B*NCHUNK*HID floats
    float* S     = (float*)d_ws;
    float* Carry = S + (size_t)NBATCH * NCHUNK * HID;

    ssm_chunk_kernel<<<dim3(NBATCH * NCHUNK), dim3(32), 0, stream>>>(
        u, A, W_B, b_B, W_C, b_C, y, S);
    ssm_carry_kernel<<<dim3(NBATCH), dim3(32), 0, stream>>>(A, S, Carry);
    ssm_fix_kernel<<<dim3(NBATCH * NCHUNK), dim3(CHLEN), 0, stream>>>(
        A, W_C, Carry, y);
}